// EnhancementLayerMamba_34497177321664
// MI455X (gfx1250) — compile-verified
//
#include <hip/hip_runtime.h>
#include <hip/hip_bf16.h>
#include <math.h>

// ---------------------------------------------------------------------------
// EnhancementLayerMamba for MI455X (gfx1250, wave32, WMMA).
//
// Key simplification: seq-len L == 1, so the selective scan collapses to
//   y[b,e] = u[b,e] * (softplus(dr@W_dt + b_dt)[b,e] * sum_n(B[b,n]C[b,n]) + D[e])
// and A_log / exp(delta*A) are dead. The layer is 7 skinny fp32 GEMMs
// (M=128) -> HBM-bandwidth bound on ~46MB of weights. We therefore use
// V_WMMA_F32_16X16X4_F32 (full fp32 precision, free at this arithmetic
// intensity), read every weight once, and share each 16-wide weight panel
// across all 8 M-tiles of a workgroup via LDS.
// ---------------------------------------------------------------------------

typedef float v2f __attribute__((ext_vector_type(2)));
typedef float v8f __attribute__((ext_vector_type(8)));

#define BATCH   128
#define STEPS   1024
#define EDIM    2048      // E = 2*STEPS
#define DTRANK  512
#define NSTATE  512       // N
#define CONDF   512

__device__ __forceinline__ float silu_f(float x)     { return x / (1.f + expf(-x)); }
__device__ __forceinline__ float softplus_f(float x) { return (x > 20.f) ? x : log1pf(expf(x)); }
__device__ __forceinline__ float gelu_f(float x)     { return 0.5f * x * (1.f + erff(x * 0.70710678118654752440f)); }

// ---------------------------------------------------------------------------
// WMMA GEMM: OUT(128 x N) = A(128 x K, row stride lda) * W(K x N, row-major)
// Block: 256 threads = 8 waves; wave w owns rows [16w, 16w+16); block owns
// 16 output columns (n0 = 16*blockIdx.x). K-chunks of 64 staged in LDS for
// the weight panel (read once from HBM, reused by all 8 waves).
// MODE selects the fused epilogue.
// ---------------------------------------------------------------------------
template<int MODE>
__global__ __launch_bounds__(256) void gemm_wmma(
    const float* __restrict__ A, int lda,
    const float* __restrict__ W, int K, int N,
    float* __restrict__ OUT,
    const float* __restrict__ aux0, const float* __restrict__ aux1,
    const float* __restrict__ aux2, const float* __restrict__ aux3,
    const float* __restrict__ aux4,
    float* __restrict__ out2)
{
    // [16 cols][64 K rows + pad 4] : pad keeps ds_load_b64 8B-aligned and the
    // two half-waves on disjoint bank groups (4c vs 4c+2 mod 64).
    __shared__ float ldsB[16][68];

    const int tid   = threadIdx.x;
    const int lane  = tid & 31;
    const int wave  = tid >> 5;          // 0..7 -> M tile
    const int mbase = wave * 16;
    const int n0    = blockIdx.x * 16;
    const int c     = lane & 15;         // column-in-tile / row-in-tile
    const int half  = lane >> 4;         // 0: K=0,1   1: K=2,3

    v8f acc = {0.f,0.f,0.f,0.f,0.f,0.f,0.f,0.f};

    const float* arow = A + (mbase + c) * lda;   // A-fragment source row

    for (int k0 = 0; k0 < K; k0 += 64) {
        __syncthreads();
        {   // cooperative load of W[k0:k0+64, n0:n0+16] into LDS (coalesced 64B rows)
            const int cc = tid & 15;
            const int r0 = tid >> 4;     // 0..15
            #pragma unroll
            for (int i = 0; i < 4; ++i) {
                const int r = r0 + i * 16;
                ldsB[cc][r] = W[(k0 + r) * N + (n0 + cc)];
            }
        }
        __syncthreads();

        #pragma unroll
        for (int kk = 0; kk < 64; kk += 4) {
            // A 16x4 f32 frag: lanes0-15 hold K=kk,kk+1; lanes16-31 K=kk+2,kk+3
            v2f afrag = *(const v2f*)(arow + k0 + kk + 2 * half);
            // B 4x16 frag, mirrored layout, from LDS
            v2f bfrag = *(const v2f*)(&ldsB[c][kk + 2 * half]);
            acc = __builtin_amdgcn_wmma_f32_16x16x4_f32(
                      false, afrag, false, bfrag, (short)0, acc, false, false);
        }
    }

    // Epilogue: C/D layout -> VGPR r: lanes0-15 row mbase+r, lanes16-31 row mbase+8+r
    #pragma unroll
    for (int r = 0; r < 8; ++r) {
        const int m   = mbase + r + 8 * half;
        const int col = n0 + c;
        float v = acc[r];

        if (MODE == 0) {
            // h@W_in -> [xi | res]; L=1 causal depthwise conv == tap3 scale+bias
            if (col < EDIM) {
                float u = v * aux0[col] + aux1[col];        // aux0=conv_w[3,0,:], aux1=conv_b
                OUT[m * EDIM + col] = silu_f(u);            // U
            } else {
                out2[m * EDIM + (col - EDIM)] = silu_f(v);  // SRES = silu(res)
            }
        } else if (MODE == 1) {
            OUT[m * N + col] = v;                           // x_dbl
        } else if (MODE == 2) {
            // delta=softplus(.+b_dt); y = u*(delta*s + D)*silu(res)
            float delta = softplus_f(v + aux0[col]);        // aux0=b_dt
            float y = (delta * aux1[m] + aux2[col])         // aux1=S[b], aux2=D
                      * aux3[m * EDIM + col]                // aux3=U
                      * aux4[m * EDIM + col];               // aux4=SRES
            OUT[m * N + col] = y;                           // YG
        } else if (MODE == 3) {
            OUT[m * N + col] = v + aux0[col] + aux1[m * STEPS + col];  // +b_out +x
        } else if (MODE == 4) {
            OUT[m * N + col] = v + aux0[col];               // gb = c@W_f + b_f
        } else if (MODE == 5) {
            // z = gelu(.+b_d);  FiLM: z*g + b  (g=gb[:, :512], b=gb[:, 512:])
            float t = gelu_f(v + aux0[col]);                // aux0=b_d
            OUT[m * N + col] = t * aux1[m * 1024 + col] + aux1[m * 1024 + col + NSTATE];
        } else { // MODE == 6
            OUT[m * N + col] = v + aux0[col];               // final: +b_o
        }
    }
}

// ---------------------------------------------------------------------------
// LayerNorm over last dim (1024) : one block per batch row.
// ---------------------------------------------------------------------------
__global__ __launch_bounds__(256) void ln_kernel(
    const float* __restrict__ x, const float* __restrict__ g,
    const float* __restrict__ b, float* __restrict__ h)
{
    __shared__ float red[2][8];
    const int row = blockIdx.x, tid = threadIdx.x;
    const float* xr = x + row * STEPS;

    float s = 0.f, s2 = 0.f;
    #pragma unroll
    for (int i = 0; i < 4; ++i) { float v = xr[tid + 256 * i]; s += v; s2 += v * v; }
    #pragma unroll
    for (int off = 16; off > 0; off >>= 1) {
        s  += __shfl_down(s,  off, 32);
        s2 += __shfl_down(s2, off, 32);
    }
    if ((tid & 31) == 0) { red[0][tid >> 5] = s; red[1][tid >> 5] = s2; }
    __syncthreads();
    if (tid == 0) {
        float ts = 0.f, ts2 = 0.f;
        #pragma unroll
        for (int w = 0; w < 8; ++w) { ts += red[0][w]; ts2 += red[1][w]; }
        red[0][0] = ts; red[1][0] = ts2;
    }
    __syncthreads();
    const float mean = red[0][0] * (1.f / STEPS);
    const float var  = red[1][0] * (1.f / STEPS) - mean * mean;
    const float inv  = rsqrtf(var + 1e-5f);
    #pragma unroll
    for (int i = 0; i < 4; ++i) {
        const int cx = tid + 256 * i;
        h[row * STEPS + cx] = (xr[cx] - mean) * inv * g[cx] + b[cx];
    }
}

// ---------------------------------------------------------------------------
// S[b] = sum_n Bm[b,n]*Cm[b,n]  (Bm = xdbl[:,512:1024], Cm = xdbl[:,1024:1536])
// ---------------------------------------------------------------------------
__global__ __launch_bounds__(256) void sdot_kernel(
    const float* __restrict__ xdbl, float* __restrict__ S)
{
    __shared__ float red[8];
    const int row = blockIdx.x, tid = threadIdx.x;
    const float* base = xdbl + row * (DTRANK + 2 * NSTATE);
    float s = base[DTRANK + tid]       * base[DTRANK + NSTATE + tid]
            + base[DTRANK + tid + 256] * base[DTRANK + NSTATE + tid + 256];
    #pragma unroll
    for (int off = 16; off > 0; off >>= 1) s += __shfl_down(s, off, 32);
    if ((tid & 31) == 0) red[tid >> 5] = s;
    __syncthreads();
    if (tid == 0) {
        float t = 0.f;
        #pragma unroll
        for (int w = 0; w < 8; ++w) t += red[w];
        S[row] = t;
    }
}

// ---------------------------------------------------------------------------
// Workspace layout (floats)
// ---------------------------------------------------------------------------
#define OFF_H     0u              // 128*1024
#define OFF_U     131072u         // 128*2048
#define OFF_SRES  393216u         // 128*2048
#define OFF_XDBL  655360u         // 128*1536
#define OFF_S     851968u         // 128
#define OFF_YG    852096u         // 128*2048
#define OFF_MOUT  1114240u        // 128*1024
#define OFF_GB    1245312u        // 128*1024
#define OFF_Z     1376384u        // 128*512

extern "C" void kernel_launch(void* const* d_in, const int* in_sizes, int n_in,
                              void* d_out, int out_size, void* d_ws, size_t ws_size,
                              hipStream_t stream) {
    const float* x      = (const float*)d_in[0];
    const float* cvec   = (const float*)d_in[1];
    const float* ln_g   = (const float*)d_in[2];
    const float* ln_b   = (const float*)d_in[3];
    const float* W_in   = (const float*)d_in[4];
    const float* conv_w = (const float*)d_in[5];   // (4,1,2048); tap 3 at +3*2048
    const float* conv_b = (const float*)d_in[6];
    const float* W_x    = (const float*)d_in[7];
    const float* W_dt   = (const float*)d_in[8];
    const float* b_dt   = (const float*)d_in[9];
    /* d_in[10] = A_log : dead code for L==1 */
    const float* Dvec   = (const float*)d_in[11];
    const float* W_out  = (const float*)d_in[12];
    const float* b_out  = (const float*)d_in[13];
    const float* W_d    = (const float*)d_in[14];
    const float* b_d    = (const float*)d_in[15];
    const float* W_f    = (const float*)d_in[16];
    const float* b_f    = (const float*)d_in[17];
    const float* W_o    = (const float*)d_in[18];
    const float* b_o    = (const float*)d_in[19];

    float* ws   = (float*)d_ws;
    float* H    = ws + OFF_H;
    float* U    = ws + OFF_U;
    float* SRES = ws + OFF_SRES;
    float* XDBL = ws + OFF_XDBL;
    float* S    = ws + OFF_S;
    float* YG   = ws + OFF_YG;
    float* MOUT = ws + OFF_MOUT;
    float* GB   = ws + OFF_GB;
    float* Z    = ws + OFF_Z;
    float* OUTP = (float*)d_out;

    const dim3 blk(256);

    // 1) LayerNorm
    ln_kernel<<<dim3(BATCH), blk, 0, stream>>>(x, ln_g, ln_b, H);

    // 2) xz = h@W_in ; fused: U = silu(xi*cw3 + cb), SRES = silu(res)
    gemm_wmma<0><<<dim3(2 * EDIM / 16), blk, 0, stream>>>(
        H, STEPS, W_in, STEPS, 2 * EDIM, U,
        conv_w + 3 * EDIM, conv_b, nullptr, nullptr, nullptr, SRES);

    // 3) FiLM gains (independent of main chain): GB = c@W_f + b_f
    gemm_wmma<4><<<dim3(2 * NSTATE / 16), blk, 0, stream>>>(
        cvec, CONDF, W_f, CONDF, 2 * NSTATE, GB,
        b_f, nullptr, nullptr, nullptr, nullptr, nullptr);

    // 4) x_dbl = u@W_x
    gemm_wmma<1><<<dim3((DTRANK + 2 * NSTATE) / 16), blk, 0, stream>>>(
        U, EDIM, W_x, EDIM, DTRANK + 2 * NSTATE, XDBL,
        nullptr, nullptr, nullptr, nullptr, nullptr, nullptr);

    // 5) S[b] = sum_n B*C   (scan collapse, L==1)
    sdot_kernel<<<dim3(BATCH), blk, 0, stream>>>(XDBL, S);

    // 6) delta = softplus(d_r@W_dt + b_dt); YG = u*(delta*S + D)*silu(res)
    gemm_wmma<2><<<dim3(EDIM / 16), blk, 0, stream>>>(
        XDBL, DTRANK + 2 * NSTATE, W_dt, DTRANK, EDIM, YG,
        b_dt, S, Dvec, U, SRES, nullptr);

    // 7) mamba_out = YG@W_out + b_out + x
    gemm_wmma<3><<<dim3(STEPS / 16), blk, 0, stream>>>(
        YG, EDIM, W_out, EDIM, STEPS, MOUT,
        b_out, x, nullptr, nullptr, nullptr, nullptr);

    // 8) z = gelu(mamba_out@W_d + b_d) * g + b   (FiLM fused)
    gemm_wmma<5><<<dim3(NSTATE / 16), blk, 0, stream>>>(
        MOUT, STEPS, W_d, STEPS, NSTATE, Z,
        b_d, GB, nullptr, nullptr, nullptr, nullptr);

    // 9) out = z@W_o + b_o
    gemm_wmma<6><<<dim3(STEPS / 16), blk, 0, stream>>>(
        Z, NSTATE, W_o, NSTATE, STEPS, OUTP,
        b_o, nullptr, nullptr, nullptr, nullptr, nullptr);
}